// MultiObjectPointMatchingLoss_59450937312080
// MI455X (gfx1250) — compile-verified
//
#include <hip/hip_runtime.h>

typedef float v2f __attribute__((ext_vector_type(2)));
typedef float v8f __attribute__((ext_vector_type(8)));

#define N_PTS   2048
#define BATCH   32
#define TILE    16
#define N_TILES (N_PTS / TILE)      /* 128 */
#define THREADS 256
#define WAVES   8                   /* 256 / wave32 */
#define GROUPS  8
#define TILES_PER_GROUP (N_TILES / GROUPS)        /* 16 */
#define TILES_PER_WAVE  (TILES_PER_GROUP / WAVES) /* 2  */

__device__ __forceinline__ void quat_to_mat(const float* q4, float R[9]) {
    float w = q4[0], x = q4[1], y = q4[2], z = q4[3];
    float n = sqrtf(w * w + x * x + y * y + z * z) + 1e-8f;
    float inv = 1.0f / n;
    w *= inv; x *= inv; y *= inv; z *= inv;
    float x2 = x * x, y2 = y * y, z2 = z * z;
    float xy = x * y, xz = x * z, yz = y * z;
    float wx = w * x, wy = w * y, wz = w * z;
    R[0] = 1.f - 2.f * (y2 + z2); R[1] = 2.f * (xy - wz);      R[2] = 2.f * (xz + wy);
    R[3] = 2.f * (xy + wz);       R[4] = 1.f - 2.f * (x2 + z2); R[5] = 2.f * (yz - wx);
    R[6] = 2.f * (xz - wy);       R[7] = 2.f * (yz + wx);      R[8] = 1.f - 2.f * (x2 + y2);
}

__global__ void pml_zero_ws(float* ws) {
    int i = threadIdx.x;
    if (i < 2 * BATCH) ws[i] = 0.0f;
}

// grid = (BATCH, GROUPS), block = 256 threads (8 wave32 waves)
__global__ __launch_bounds__(THREADS)
void pml_adds_kernel(const float* __restrict__ pred_q, const float* __restrict__ pred_t,
                     const float* __restrict__ gt_q,   const float* __restrict__ gt_t,
                     const float* __restrict__ point_bank, const int* __restrict__ cls,
                     float* __restrict__ add_sum, float* __restrict__ adds_sum) {
    const int b    = blockIdx.x;
    const int grp  = blockIdx.y;
    const int tid  = threadIdx.x;
    const int lane = tid & 31;
    const int wave = tid >> 5;
    const int half = lane >> 4;   // 0: lanes 0-15, 1: lanes 16-31
    const int l16  = lane & 15;

    // B-fragment staged layout:
    //   bfrag[0][i] = (gx, gy)   consumed by lanes 0-15  (K=0,1)
    //   bfrag[1][i] = (gz, g^2)  consumed by lanes 16-31 (K=2,3)
    __shared__ float2 bfrag[2][N_PTS]; // 32 KB

    const int c = cls[b];
    const float* pts = point_bank + (size_t)c * N_PTS * 3;

    float Rp[9], Rg[9];
    quat_to_mat(pred_q + b * 4, Rp);
    quat_to_mat(gt_q + b * 4, Rg);
    const float tpx = pred_t[b * 3 + 0], tpy = pred_t[b * 3 + 1], tpz = pred_t[b * 3 + 2];
    const float tgx = gt_t[b * 3 + 0],   tgy = gt_t[b * 3 + 1],   tgz = gt_t[b * 3 + 2];

    // Stage transformed gt points into LDS; group 0 also accumulates the ADD loss.
    float addAcc = 0.0f;
    for (int i = tid; i < N_PTS; i += THREADS) {
        float px = pts[i * 3 + 0], py = pts[i * 3 + 1], pz = pts[i * 3 + 2];
        float gx = Rg[0] * px + Rg[1] * py + Rg[2] * pz + tgx;
        float gy = Rg[3] * px + Rg[4] * py + Rg[5] * pz + tgy;
        float gz = Rg[6] * px + Rg[7] * py + Rg[8] * pz + tgz;
        float g2 = gx * gx + gy * gy + gz * gz;
        bfrag[0][i] = make_float2(gx, gy);
        bfrag[1][i] = make_float2(gz, g2);
        if (grp == 0) {
            float qx = Rp[0] * px + Rp[1] * py + Rp[2] * pz + tpx;
            float qy = Rp[3] * px + Rp[4] * py + Rp[5] * pz + tpy;
            float qz = Rp[6] * px + Rp[7] * py + Rp[8] * pz + tpz;
            float dx = qx - gx, dy = qy - gy, dz = qz - gz;
            addAcc += sqrtf(dx * dx + dy * dy + dz * dz);
        }
    }
    if (grp == 0) {
        for (int s = 16; s > 0; s >>= 1) addAcc += __shfl_xor(addAcc, s, 32);
        if (lane == 0) atomicAdd(&add_sum[b], addAcc);
    }
    __syncthreads();

    // Per-lane B base pointer: resolves the half-wave select once, outside the hot loop.
    const float2* __restrict__ bbase = &bfrag[half][0];

    float addsAcc = 0.0f;
    for (int t = 0; t < TILES_PER_WAVE; ++t) {
        const int rowTile = grp * TILES_PER_GROUP + t * WAVES + wave;
        const int row = rowTile * TILE + l16;

        // transform pred point for this lane's row
        float px = pts[row * 3 + 0], py = pts[row * 3 + 1], pz = pts[row * 3 + 2];
        float ppx = Rp[0] * px + Rp[1] * py + Rp[2] * pz + tpx;
        float ppy = Rp[3] * px + Rp[4] * py + Rp[5] * pz + tpy;
        float ppz = Rp[6] * px + Rp[7] * py + Rp[8] * pz + tpz;
        float p2  = ppx * ppx + ppy * ppy + ppz * ppz;

        // A fragment (16x4 f32): lanes 0-15 carry K=0,1 ; lanes 16-31 carry K=2,3
        v2f a;
        a.x = half ? (-2.0f * ppz) : (-2.0f * ppx);
        a.y = half ? 1.0f          : (-2.0f * ppy);

        // C init = p^2 per row: VGPR v holds row (v + 8*half); row m's p2 lives in lane m
        v8f cinit;
#pragma unroll
        for (int v = 0; v < 8; ++v) cinit[v] = __shfl(p2, v + half * 8, 32);

        v8f dmin;
#pragma unroll
        for (int v = 0; v < 8; ++v) dmin[v] = 3.4e38f;

#pragma unroll 4
        for (int ct = 0; ct < N_TILES; ++ct) {
            float2 g = bbase[ct * TILE + l16];  // ds_load_b64, no selects
            v2f bb;
            bb.x = g.x;
            bb.y = g.y;
            // D[i][j] = p2_i + g2_j - 2 p_i . g_j  == squared pairwise distance
            v8f d = __builtin_amdgcn_wmma_f32_16x16x4_f32(
                false, a, false, bb, (short)0, cinit, false, false);
#pragma unroll
            for (int v = 0; v < 8; ++v) dmin[v] = fminf(dmin[v], d[v]);
        }

        // min over the 16 columns held across each 16-lane half
#pragma unroll
        for (int v = 0; v < 8; ++v) {
            float m = dmin[v];
            m = fminf(m, __shfl_xor(m, 1, 32));
            m = fminf(m, __shfl_xor(m, 2, 32));
            m = fminf(m, __shfl_xor(m, 4, 32));
            m = fminf(m, __shfl_xor(m, 8, 32));
            dmin[v] = m;
        }
        if (l16 == 0) {
            float s = 0.0f;
#pragma unroll
            for (int v = 0; v < 8; ++v) s += sqrtf(fmaxf(dmin[v], 0.0f));
            addsAcc += s;  // lane 0: rows 0-7, lane 16: rows 8-15 of each tile
        }
    }
    if (l16 == 0) atomicAdd(&adds_sum[b], addsAcc);
}

__global__ void pml_finalize(const float* __restrict__ add_sum,
                             const float* __restrict__ adds_sum,
                             const int* __restrict__ cls, float* __restrict__ out) {
    int b = threadIdx.x; // 32 threads, one wave
    int c = cls[b];
    bool sym = (c == 10) || (c == 11);
    float loss = (sym ? adds_sum[b] : add_sum[b]) * (1.0f / (float)N_PTS);
    for (int s = 16; s > 0; s >>= 1) loss += __shfl_xor(loss, s, 32);
    if (b == 0) out[0] = loss * (1.0f / (float)BATCH);
}

extern "C" void kernel_launch(void* const* d_in, const int* in_sizes, int n_in,
                              void* d_out, int out_size, void* d_ws, size_t ws_size,
                              hipStream_t stream) {
    const float* pred_q = (const float*)d_in[0];
    const float* pred_t = (const float*)d_in[1];
    const float* gt_q   = (const float*)d_in[2];
    const float* gt_t   = (const float*)d_in[3];
    const float* bank   = (const float*)d_in[4];
    const int*   cls    = (const int*)d_in[5];
    float* out = (float*)d_out;
    float* ws  = (float*)d_ws;          // [0,32): add_sum, [32,64): adds_sum

    pml_zero_ws<<<1, 64, 0, stream>>>(ws);
    dim3 grid(BATCH, GROUPS);
    pml_adds_kernel<<<grid, THREADS, 0, stream>>>(pred_q, pred_t, gt_q, gt_t,
                                                  bank, cls, ws, ws + BATCH);
    pml_finalize<<<1, 32, 0, stream>>>(ws, ws + BATCH, cls, out);
}